// multi_SpatialEmbLoss_15315853377952
// MI455X (gfx1250) — compile-verified
//
#include <hip/hip_runtime.h>
#include <math.h>

// Problem constants (from reference)
#define Bn 8
#define Hn 512
#define Wn 512
#define HW (Hn*Wn)
#define NIDS 15          // ids 1..15
#define NBINS 256        // error histogram bins over [0,2]

// Workspace layout (in floats)
#define OFF_STATS   0                         // [B][NIDS][5]: cnt,sumx,sumy,sumsig,sumsig2 (atomic)
#define OFF_DERIV   600                       // [B][NIDS][4]: cx,cy,s,var
#define OFF_BGSEED  1080                      // [B] (atomic)
#define OFF_FOCAL   1088                      // [1] (atomic)
#define OFF_SEEDFG  1092                      // [B][NIDS] (atomic)
#define OFF_LOV     1212                      // [B][NIDS]
#define OFF_HIST    1536                      // [B][3][NIDS*NBINS] fg/bg/errsum (atomic)
#define HIST_PER_B  (3*NIDS*NBINS)
#define WS_FLOATS   (OFF_HIST + Bn*HIST_PER_B)

typedef __attribute__((ext_vector_type(2))) float v2f;
typedef __attribute__((ext_vector_type(8))) float v8f;

// Fast transcendental helpers: one v_exp_f32 + one v_rcp_f32 each,
// avoiding the IEEE div (div_scale/div_fmas/div_fixup) sequence.
__device__ __forceinline__ float fast_sigmoid(float x){
    return __builtin_amdgcn_rcpf(1.0f + __expf(-x));
}
__device__ __forceinline__ float fast_tanh(float x){
    // tanh(x) = 1 - 2/(exp(2x)+1)
    return 1.0f - 2.0f*__builtin_amdgcn_rcpf(__expf(2.0f*x) + 1.0f);
}

// ---------------------------------------------------------------------------
// Stage 1: per-(b,k) mask stats, bg seed, focal loss. LDS accumulate + flush.
// ---------------------------------------------------------------------------
__global__ void k_stats(const float* __restrict__ pred, const int* __restrict__ labels,
                        const int* __restrict__ inst, float* __restrict__ ws)
{
    int b = blockIdx.y;
    int p = blockIdx.x*blockDim.x + threadIdx.x;   // pixel index in [0,HW)
    int tid = threadIdx.x;
    __shared__ float acc[NIDS*5 + 2];              // stats + bgseed + focal
    for (int i = tid; i < NIDS*5 + 2; i += blockDim.x) acc[i] = 0.f;
    __syncthreads();

    int y = p >> 9, x = p & 511;
    const float* pb = pred + (size_t)b*6*HW;
    float xm = x*(2.0f/511.0f);                    // linspace(0, W/RES=2, 512)
    float ym = y*(2.0f/511.0f);

    int id = inst[(size_t)b*HW + p];
    if (id >= 1 && id <= NIDS){
        float sg = pb[2*HW + p];                   // sigma channel
        float* a = &acc[(id-1)*5];
        atomicAdd(&a[0], 1.0f);
        atomicAdd(&a[1], xm);
        atomicAdd(&a[2], ym);
        atomicAdd(&a[3], sg);
        atomicAdd(&a[4], sg*sg);
    }

    float l3 = pb[3*HW+p], l4 = pb[4*HW+p], l5 = pb[5*HW+p];
    int lb0 = labels[(size_t)b*3*HW + p];
    int lb1 = labels[(size_t)b*3*HW + HW + p];
    int lb2 = labels[(size_t)b*3*HW + 2*HW + p];

    if (lb2 == 0){                                 // background seed term
        float sd = fast_sigmoid(l5);
        atomicAdd(&acc[NIDS*5], sd*sd);
    }
    // focal loss: target = argmax(labels) (first max wins, as jnp.argmax)
    int targ = (lb0 >= lb1 && lb0 >= lb2) ? 0 : ((lb1 >= lb2) ? 1 : 2);
    float mx = fmaxf(l3, fmaxf(l4, l5));
    float lse = mx + __logf(__expf(l3-mx)+__expf(l4-mx)+__expf(l5-mx));
    float lt  = (targ==0) ? l3 : ((targ==1) ? l4 : l5);
    float lp  = lt - lse;
    float pt  = __expf(lp);
    float om  = 1.0f - pt;
    atomicAdd(&acc[NIDS*5+1], -om*om*lp);          // GAMMA = 2

    __syncthreads();
    float* gs = ws + OFF_STATS + b*NIDS*5;
    for (int i = tid; i < NIDS*5; i += blockDim.x) atomicAdd(&gs[i], acc[i]);
    if (tid == 0) atomicAdd(&ws[OFF_BGSEED + b], acc[NIDS*5]);
    if (tid == 1) atomicAdd(&ws[OFF_FOCAL],      acc[NIDS*5+1]);
}

// ---------------------------------------------------------------------------
// Stage 1.5: derive centers, s = exp(10*mean sigma), var = E[s^2]-mean^2
// ---------------------------------------------------------------------------
__global__ void k_derive(float* __restrict__ ws)
{
    int i = threadIdx.x;
    if (i >= Bn*NIDS) return;
    const float* st = ws + OFF_STATS + i*5;
    float cnt  = st[0];
    float safe = fmaxf(cnt, 1.0f);
    float cx = st[1]/safe, cy = st[2]/safe;
    float sm = st[3]/safe;
    float var = st[4]/safe - sm*sm;                // forward value of mean((s-sg(sm))^2)
    float s = __expf(10.0f*sm);
    float* dv = ws + OFF_DERIV + i*4;
    dv[0]=cx; dv[1]=cy; dv[2]=s; dv[3]=var;
}

// ---------------------------------------------------------------------------
// Stage 2: WMMA distance kernel, ids->M rows, pixels->N cols.
//   A (loop-invariant, per id m): [-2cx, -2cy, cx^2+cy^2, 0]   (16x4)
//   B (per chunk, per pixel n):   [ex, ey, 1, 0]               (4x16)
//   D[m][n] = -2 e.c + |c|^2 ; lane L owns (m = r+8*(L>>4), n = L&15),
//   so all 8 D elements of a lane share the lane's OWN pixel -> no cross-lane
//   shuffles in the epilogue. Per-id s values preloaded per r (loop-invariant).
// ---------------------------------------------------------------------------
#define WAVES  8
#define CHUNKS 64        // 8 waves * 64 chunks * 16 px = 8192 px / block

__global__ void __launch_bounds__(256) k_dist(const float* __restrict__ pred,
                                              const int* __restrict__ inst,
                                              float* __restrict__ ws)
{
    int b    = blockIdx.y;
    int tid  = threadIdx.x;
    int wv   = tid >> 5;
    int lane = tid & 31;
    int half = lane >> 4;                           // 0: K=0,1 / ids 0-7 ; 1: K=2,3 / ids 8-15
    int q    = lane & 15;                           // id row (for A) and pixel slot (for B/D)

    __shared__ float hist[3*NIDS*NBINS];            // fg, bg, errsum
    for (int i = tid; i < 3*NIDS*NBINS; i += 256) hist[i] = 0.f;
    __syncthreads();

    // A fragment: row m = q, loop-invariant
    float cx = 0.f, cy = 0.f;
    if (q >= 1){
        const float* dv = ws + OFF_DERIV + (b*NIDS + (q-1))*4;
        cx = dv[0]; cy = dv[1];
    }
    v2f am;
    if (half == 0){ am.x = -2.0f*cx;     am.y = -2.0f*cy; }   // K=0,1
    else          { am.x = cx*cx+cy*cy;  am.y = 0.0f;     }   // K=2,3

    // Per-r loop invariants: this lane's D rows are ids m_r = r + 8*half
    float sArr[8];
    #pragma unroll
    for (int r = 0; r < 8; r++){
        int m = r + 8*half;
        sArr[r] = (m >= 1) ? ws[OFF_DERIV + (b*NIDS + (m-1))*4 + 2] : 0.f;
    }

    const float* pb = pred + (size_t)b*6*HW;
    const int*   ib = inst + (size_t)b*HW;
    float sAcc[8];
    #pragma unroll
    for (int r = 0; r < 8; r++) sAcc[r] = 0.f;

    for (int c = 0; c < CHUNKS; c++){
        int p = ((blockIdx.x*WAVES + wv)*CHUNKS + c)*16 + q;   // lane's own pixel
        int y = p >> 9, x = p & 511;
        float ex   = fast_tanh(pb[p])      + x*(2.0f/511.0f);
        float ey   = fast_tanh(pb[HW + p]) + y*(2.0f/511.0f);
        float esq  = ex*ex + ey*ey;
        float seed = fast_sigmoid(pb[5*HW + p]);
        int   iv   = ib[p];

        v2f bm;
        if (half == 0){ bm.x = ex;   bm.y = ey;  }   // K=0,1 for pixel col q
        else          { bm.x = 1.0f; bm.y = 0.f; }   // K=2,3
        v8f cz = {};
        v8f d = __builtin_amdgcn_wmma_f32_16x16x4_f32(
                    false, am, false, bm, (short)0, cz, false, false);

        #pragma unroll
        for (int r = 0; r < 8; r++){
            int m = r + 8*half;                      // id of this D row
            float dist = __expf(-sArr[r]*(esq + d[r]));
            if (m >= 1){
                int   fg  = (iv == m);
                float err = fg ? (2.0f - 2.0f*dist) : (2.0f*dist);  // in [0,2]
                int bin = (int)(err * 128.0f);
                bin = bin < 0 ? 0 : (bin > NBINS-1 ? NBINS-1 : bin);
                int idx = (m-1)*NBINS + bin;
                atomicAdd(&hist[(fg ? 0 : 1)*NIDS*NBINS + idx], 1.0f);
                atomicAdd(&hist[2*NIDS*NBINS + idx], err);
                if (fg){ float t = seed - dist; sAcc[r] += t*t; }
            }
        }
    }
    #pragma unroll
    for (int r = 0; r < 8; r++){
        int m = r + 8*half;
        if (m >= 1) atomicAdd(&ws[OFF_SEEDFG + b*NIDS + (m-1)], sAcc[r]);
    }
    __syncthreads();
    float* gh = ws + OFF_HIST + b*HIST_PER_B;
    for (int i = tid; i < 3*NIDS*NBINS; i += 256) atomicAdd(&gh[i], hist[i]);
}

// ---------------------------------------------------------------------------
// Stage 3: Lovasz via histogram. Telescoped sum over equal-error runs:
// lov = sum_bins errmean * (J(cum_incl) - J(cum_excl)),  J = 1-(g-cf)/(g+cb)
// Processed in descending error order (bin NBINS-1 -> 0).
// ---------------------------------------------------------------------------
__global__ void k_lovasz(float* __restrict__ ws)
{
    int bk = blockIdx.x;                 // 0..119
    int b = bk / NIDS, k = bk % NIDS;    // id = k+1
    int t = threadIdx.x;                 // descending position
    int bin = NBINS - 1 - t;
    const float* gh = ws + OFF_HIST + b*HIST_PER_B;
    int idx = k*NBINS + bin;
    float fg = gh[idx];
    float bg = gh[NIDS*NBINS + idx];
    float es = gh[2*NIDS*NBINS + idx];

    __shared__ float sfg[NBINS], sbg[NBINS], red[NBINS];
    sfg[t] = fg; sbg[t] = bg;
    __syncthreads();
    for (int off = 1; off < NBINS; off <<= 1){     // Hillis-Steele inclusive scan
        float af = (t >= off) ? sfg[t-off] : 0.f;
        float ab = (t >= off) ? sbg[t-off] : 0.f;
        __syncthreads();
        sfg[t] += af; sbg[t] += ab;
        __syncthreads();
    }

    float gts = ws[OFF_STATS + (b*NIDS + k)*5];    // cnt
    float cfi = sfg[t], cbi = sbg[t];
    float cfe = cfi - fg, cbe = cbi - bg;
    float tot = fg + bg;
    float contrib = 0.f;
    if (tot > 0.f && gts > 0.5f){
        float g  = fmaxf(gts, 1.0f);
        float Ji = 1.f - (g - cfi)/(g + cbi);
        float Je = 1.f - (g - cfe)/(g + cbe);      // at t=0 this is exactly 0
        contrib = (es/tot)*(Ji - Je);
    }
    red[t] = contrib; __syncthreads();
    for (int off = NBINS/2; off > 0; off >>= 1){
        if (t < off) red[t] += red[t+off];
        __syncthreads();
    }
    if (t == 0) ws[OFF_LOV + bk] = (gts > 0.5f) ? red[0] : 0.f;
}

// ---------------------------------------------------------------------------
// Stage 4: combine (trivial scalar work)
// ---------------------------------------------------------------------------
__global__ void k_final(const float* __restrict__ ws, float* __restrict__ out)
{
    if (threadIdx.x != 0 || blockIdx.x != 0) return;
    float loss = 0.f;
    for (int b = 0; b < Bn; b++){
        float nv = 0.f, il = 0.f, vl = 0.f, sf = 0.f;
        for (int k = 0; k < NIDS; k++){
            float cnt = ws[OFF_STATS + (b*NIDS + k)*5];
            if (cnt > 0.5f){
                nv += 1.f;
                il += ws[OFF_LOV + b*NIDS + k];
                vl += ws[OFF_DERIV + (b*NIDS + k)*4 + 3];
            }
            sf += ws[OFF_SEEDFG + b*NIDS + k];
        }
        float nobj = fmaxf(nv, 1.f);
        float per = (il/nobj) + 10.0f*(vl/nobj)
                  + (ws[OFF_BGSEED + b] + sf)/(float)HW;
        loss += per/(float)Bn;
    }
    loss += ws[OFF_FOCAL]/(float)((size_t)Bn*HW);  // focal mean
    out[0] = loss;
}

// ---------------------------------------------------------------------------
extern "C" void kernel_launch(void* const* d_in, const int* in_sizes, int n_in,
                              void* d_out, int out_size, void* d_ws, size_t ws_size,
                              hipStream_t stream)
{
    const float* pred   = (const float*)d_in[0];
    const int*   labels = (const int*)d_in[1];
    const int*   insts  = (const int*)d_in[2];
    float* ws  = (float*)d_ws;
    float* out = (float*)d_out;

    hipMemsetAsync(d_ws, 0, (size_t)WS_FLOATS*sizeof(float), stream);

    k_stats <<<dim3(HW/256, Bn), 256, 0, stream>>>(pred, labels, insts, ws);
    k_derive<<<1, 128, 0, stream>>>(ws);
    k_dist  <<<dim3(HW/(WAVES*CHUNKS*16), Bn), 256, 0, stream>>>(pred, insts, ws);
    k_lovasz<<<Bn*NIDS, NBINS, 0, stream>>>(ws);
    k_final <<<1, 1, 0, stream>>>(ws, out);
}